// AvoidanceHOCBFLayer_67370857005318
// MI455X (gfx1250) — compile-verified
//
#include <hip/hip_runtime.h>

// AvoidanceHOCBFLayer: batched HOCBF-QP via primal-dual interior point.
// B=16384 QPs, m=54 constraints (padded to 64), n=4 vars, 40 iterations.
// Layout: 1 wave = 4 QPs. 8 lanes/QP for elementwise+reductions; the Gram
// M = A^T diag(w) A is computed with fp32 WMMA (v_wmma_f32_16x16x4_f32),
// 16 chained K=4 chunks covering 64 constraints, 4 QPs per 16x16 tile.

#define MC 64        // padded constraint count
#define NITER 40
#define BIGC 1000.0f

typedef __attribute__((ext_vector_type(2))) float v2f;
typedef __attribute__((ext_vector_type(8))) float v8f;

__device__ __forceinline__ float grp8_sum(float v) {
  v += __shfl_xor(v, 1, 32);
  v += __shfl_xor(v, 2, 32);
  v += __shfl_xor(v, 4, 32);
  return v;
}
__device__ __forceinline__ float grp8_min(float v) {
  v = fminf(v, __shfl_xor(v, 1, 32));
  v = fminf(v, __shfl_xor(v, 2, 32));
  v = fminf(v, __shfl_xor(v, 4, 32));
  return v;
}

__global__ __launch_bounds__(256) void hocbf_ipm_kernel(
    const float* __restrict__ u_nominal, const float* __restrict__ v_current,
    const float* __restrict__ p_obs, const int* __restrict__ obs_mask,
    const float* __restrict__ p_agents, const float* __restrict__ v_agents_local,
    const int* __restrict__ agents_mask, float* __restrict__ out, int Bn)
{
  // Per-wave LDS regions (8 waves / block). [wave][component][qp-slot][constraint]
  __shared__ float lds_g[8][4][4][MC];  // static constraint rows A[i][n]
  __shared__ float lds_w[8][4][MC];     // per-iteration w = lam/s
  __shared__ float lds_M[8][4][16];     // 4x4 Gram result per qp-slot

  const int tid  = threadIdx.x;
  const int wid  = tid >> 5;
  const int lane = tid & 31;
  const int slot = lane >> 3;   // QP slot within wave (0..3)
  const int sub  = lane & 7;    // lane within QP group (0..7)
  const int raw_batch = blockIdx.x * 32 + wid * 4 + slot;
  const int batch = raw_batch < Bn ? raw_batch : (Bn - 1);  // clamp: keep EXEC full

  // Per-lane constraint registers: 8 constraints, i = sub*8 + j
  float g0[8], g1[8], g2[8], g3[8], bb[8], sv[8], lv[8];

  const float v = v_current[batch];

  #pragma unroll
  for (int j = 0; j < 8; ++j) {
    const int i = sub * 8 + j;
    float G0 = 0.f, G1 = 0.f, G2 = 0.f, G3 = 0.f, Bi = BIGC;
    if (i == 0)      { G2 = -1.f; Bi = 0.f; }                       // slack do >= 0
    else if (i == 1) { G3 = -1.f; Bi = 0.f; }                       // slack da >= 0
    else if (i < 34) {                                              // obstacle rows
      const int o = i - 2;
      const float lx = p_obs[(batch * 32 + o) * 2 + 0];
      const float ly = p_obs[(batch * 32 + o) * 2 + 1];
      if (obs_mask[batch * 32 + o]) {
        G0 = 2.f * lx; G1 = 2.f * ly * v; G2 = -1.f;
        const float h = lx * lx + ly * ly - 0.25f;                  // D_OBS^2
        Bi = 2.f * v * v + 3.f * (-2.f * lx * v) + 2.f * h;         // DAMP=3, STIFF=2
      }
    }
    else if (i < 49) {                                              // agent rows
      const int a = i - 34;
      const float ax  = p_agents[(batch * 15 + a) * 2 + 0];
      const float ay  = p_agents[(batch * 15 + a) * 2 + 1];
      const float vjx = v_agents_local[(batch * 15 + a) * 2 + 0];
      const float vjy = v_agents_local[(batch * 15 + a) * 2 + 1];
      if (agents_mask[batch * 15 + a]) {
        G0 = 2.f * ax;
        G1 = 2.f * ay * v - 2.f * ay * vjx + 2.f * ax * vjy;
        G3 = -1.f;
        const float h   = ax * ax + ay * ay - 1.f;                  // D_SAFE^2
        const float hd  = -2.f * ax * v + 2.f * (ax * vjx + ay * vjy);
        const float hdd = 2.f * v * v - 4.f * v * vjx + 2.f * vjx * vjx + 2.f * vjy * vjy;
        Bi = hdd + 3.f * hd + 2.f * h;                              // DAMP_A=3, STIFF_A=2
      }
    }
    else if (i >= 50 && i < 54) {                                   // box rows
      const int k = i - 50;
      G0 = (k == 0) ? -1.f : (k == 1) ? 1.f : 0.f;
      G1 = (k == 2) ? -1.f : (k == 3) ? 1.f : 0.f;
      Bi = 1.f;                                                     // A_MAX = W_MAX = 1
    }
    // i==49 and i>=54: inert zero rows, b = BIG (pads are exact copies of row 49)
    g0[j] = G0; g1[j] = G1; g2[j] = G2; g3[j] = G3; bb[j] = Bi;
    sv[j] = fmaxf(Bi, 1.f);  // s0 = max(b, 1)
    lv[j] = 1.f;             // lam0 = 1
    lds_g[wid][0][slot][i] = G0;
    lds_g[wid][1][slot][i] = G1;
    lds_g[wid][2][slot][i] = G2;
    lds_g[wid][3][slot][i] = G3;
  }

  const float c0 = u_nominal[batch * 2 + 0];   // action_scale = [1,1]
  const float c1 = u_nominal[batch * 2 + 1];
  float x0 = 0.f, x1 = 0.f, x2 = 0.f, x3 = 0.f;

  // WMMA feed mapping (assumed B 4x16 layout mirrors A: VGPR0 holds K=2*khalf).
  // A[m=4b+r][k] = w_bk * g_b[k][r]; B[k][n=4b+c] = g_b[k][c]
  // => diagonal 4x4 blocks of D are M_b = A_b^T diag(w_b) A_b.
  const int wq    = (lane & 15) >> 2;  // QP slot this lane feeds
  const int wr    = lane & 3;          // matrix row/col component
  const int khalf = lane >> 4;
  const float* gRow = &lds_g[wid][wr][wq][0];
  const float* wRow = &lds_w[wid][wq][0];

  asm volatile("s_wait_dscnt 0" ::: "memory");  // fence: lds_g visible to wmma phase

  for (int it = 0; it < NITER; ++it) {
    // ---- r_d = 2*pdiag*x - 2c + A^T lam
    float a0 = 0.f, a1 = 0.f, a2 = 0.f, a3 = 0.f;
    #pragma unroll
    for (int j = 0; j < 8; ++j) {
      a0 = fmaf(g0[j], lv[j], a0);
      a1 = fmaf(g1[j], lv[j], a1);
      a2 = fmaf(g2[j], lv[j], a2);
      a3 = fmaf(g3[j], lv[j], a3);
    }
    a0 = grp8_sum(a0); a1 = grp8_sum(a1); a2 = grp8_sum(a2); a3 = grp8_sum(a3);
    const float rd0 = 2.f * x0 - 2.f * c0 + a0;
    const float rd1 = 2.f * x1 - 2.f * c1 + a1;
    const float rd2 = 2000.f * x2 + a2;          // pdiag = [1,1,1000,1000], c2=c3=0
    const float rd3 = 2000.f * x3 + a3;

    // ---- r_p = A x + s - b ; mu = 0.1 * mean(s*lam) over the 54 real rows
    float rp[8]; float msum = 0.f;
    #pragma unroll
    for (int j = 0; j < 8; ++j) {
      const int i = sub * 8 + j;
      const float axv = g0[j] * x0 + g1[j] * x1 + g2[j] * x2 + g3[j] * x3;
      rp[j] = axv + sv[j] - bb[j];
      if (i < 54) msum += sv[j] * lv[j];
    }
    const float mu = 0.1f * (grp8_sum(msum) * (1.f / 54.f));

    // ---- w = lam/s (to LDS for WMMA); rhs = -r_d - A^T t, t = (mu - s*lam + lam*r_p)/s
    float t0 = 0.f, t1 = 0.f, t2 = 0.f, t3 = 0.f;
    #pragma unroll
    for (int j = 0; j < 8; ++j) {
      const int i = sub * 8 + j;
      const float wv = lv[j] / sv[j];
      lds_w[wid][slot][i] = wv;
      const float tt = (mu - sv[j] * lv[j] + lv[j] * rp[j]) / sv[j];
      t0 = fmaf(g0[j], tt, t0);
      t1 = fmaf(g1[j], tt, t1);
      t2 = fmaf(g2[j], tt, t2);
      t3 = fmaf(g3[j], tt, t3);
    }
    t0 = grp8_sum(t0); t1 = grp8_sum(t1); t2 = grp8_sum(t2); t3 = grp8_sum(t3);
    const float rhs0 = -rd0 - t0;
    const float rhs1 = -rd1 - t1;
    const float rhs2 = -rd2 - t2;
    const float rhs3 = -rd3 - t3;

    asm volatile("s_wait_dscnt 0" ::: "memory");  // w written (same wave, in-order LDS)

    // ---- Gram: M_b = A_b^T diag(w_b) A_b via 16 x v_wmma_f32_16x16x4_f32
    v8f acc = {0.f, 0.f, 0.f, 0.f, 0.f, 0.f, 0.f, 0.f};
    #pragma unroll
    for (int ch = 0; ch < 16; ++ch) {
      const int k0 = ch * 4 + khalf * 2;
      const float gg0 = gRow[k0], gg1 = gRow[k0 + 1];
      const float ww0 = wRow[k0], ww1 = wRow[k0 + 1];
      v2f Aop; Aop.x = ww0 * gg0; Aop.y = ww1 * gg1;
      v2f Bop; Bop.x = gg0;       Bop.y = gg1;
      acc = __builtin_amdgcn_wmma_f32_16x16x4_f32(
          false, Aop, false, Bop, (short)0, acc, false, false);
    }

    // ---- scatter the 4 diagonal 4x4 blocks of D into lds_M
    // D layout: VGPR j, lanes 0-15 -> D[j][lane]; lanes 16-31 -> D[8+j][lane-16]
    if (lane < 8 || lane >= 24) {
      const int flag = (lane >> 2) & 1;
      const int qb   = (lane < 8) ? flag : (2 + flag);
      const int cc   = lane & 3;
      #pragma unroll
      for (int rr = 0; rr < 4; ++rr) {
        const float val = flag ? acc[rr + 4] : acc[rr];
        lds_M[wid][qb][rr * 4 + cc] = val;
      }
    }
    asm volatile("s_wait_dscnt 0" ::: "memory");

    // ---- read M (+ 2*pdiag on diag) and solve via Cholesky (SPD; M23==0 naturally)
    const float* Mm = &lds_M[wid][slot][0];
    const float m00 = Mm[0] + 2.f;
    const float m10 = Mm[4],  m11 = Mm[5] + 2.f;
    const float m20 = Mm[8],  m21 = Mm[9],  m22 = Mm[10] + 2000.f;
    const float m30 = Mm[12], m31 = Mm[13], m32 = Mm[14], m33 = Mm[15] + 2000.f;

    const float L00 = sqrtf(m00),               iL00 = 1.f / L00;
    const float L10 = m10 * iL00, L20 = m20 * iL00, L30 = m30 * iL00;
    const float L11 = sqrtf(m11 - L10 * L10),   iL11 = 1.f / L11;
    const float L21 = (m21 - L20 * L10) * iL11;
    const float L31 = (m31 - L30 * L10) * iL11;
    const float L22 = sqrtf(m22 - L20 * L20 - L21 * L21), iL22 = 1.f / L22;
    const float L32 = (m32 - L30 * L20 - L31 * L21) * iL22;
    const float L33 = sqrtf(m33 - L30 * L30 - L31 * L31 - L32 * L32), iL33 = 1.f / L33;

    const float y0 = rhs0 * iL00;
    const float y1 = (rhs1 - L10 * y0) * iL11;
    const float y2 = (rhs2 - L20 * y0 - L21 * y1) * iL22;
    const float y3 = (rhs3 - L30 * y0 - L31 * y1 - L32 * y2) * iL33;
    const float dx3 = y3 * iL33;
    const float dx2 = (y2 - L32 * dx3) * iL22;
    const float dx1 = (y1 - L21 * dx2 - L31 * dx3) * iL11;
    const float dx0 = (y0 - L10 * dx1 - L20 * dx2 - L30 * dx3) * iL00;

    // ---- ds, dlam, fraction-to-boundary step
    float amin_s = 1e9f, amin_l = 1e9f;
    float dsv[8], dlv[8];
    #pragma unroll
    for (int j = 0; j < 8; ++j) {
      const float adx = g0[j] * dx0 + g1[j] * dx1 + g2[j] * dx2 + g3[j] * dx3;
      const float ds_ = -rp[j] - adx;
      const float dl_ = (mu - sv[j] * lv[j] - lv[j] * ds_) / sv[j];
      dsv[j] = ds_; dlv[j] = dl_;
      if (ds_ < 0.f) amin_s = fminf(amin_s, -sv[j] / ds_);
      if (dl_ < 0.f) amin_l = fminf(amin_l, -lv[j] / dl_);
    }
    amin_s = grp8_min(amin_s);
    amin_l = grp8_min(amin_l);
    const float alpha = fminf(1.f, 0.99f * fminf(amin_s, amin_l));

    x0 += alpha * dx0; x1 += alpha * dx1; x2 += alpha * dx2; x3 += alpha * dx3;
    #pragma unroll
    for (int j = 0; j < 8; ++j) { sv[j] += alpha * dsv[j]; lv[j] += alpha * dlv[j]; }
  }

  if (sub == 0 && raw_batch < Bn) {
    out[raw_batch * 2 + 0] = x0;   // action_scale = [1,1] -> output = x[:2]
    out[raw_batch * 2 + 1] = x1;
  }
}

extern "C" void kernel_launch(void* const* d_in, const int* in_sizes, int n_in,
                              void* d_out, int out_size, void* d_ws, size_t ws_size,
                              hipStream_t stream) {
  const float* u_nominal     = (const float*)d_in[0];
  const float* v_current     = (const float*)d_in[1];
  const float* p_obs         = (const float*)d_in[2];
  const int*   obs_mask      = (const int*)d_in[3];
  const float* p_agents      = (const float*)d_in[4];
  const float* v_agents      = (const float*)d_in[5];
  const int*   agents_mask   = (const int*)d_in[6];
  float* out = (float*)d_out;

  const int Bn = in_sizes[1];              // v_current is (B,)
  const int blocks = (Bn + 31) / 32;       // 32 QPs per 256-thread block (8 waves x 4)
  hipLaunchKernelGGL(hocbf_ipm_kernel, dim3(blocks), dim3(256), 0, stream,
                     u_nominal, v_current, p_obs, obs_mask,
                     p_agents, v_agents, agents_mask, out, Bn);
}